// MultiHeadAttentionLayer_34651796144198
// MI455X (gfx1250) — compile-verified
//
#include <hip/hip_runtime.h>
#include <math.h>

#define IN_DIM   128
#define QKV_DIM  128   // N_HEADS * OUT_DIM
#define N_HEADS  8
#define HEAD_DIM 16

typedef __attribute__((ext_vector_type(2))) float v2f;
typedef __attribute__((ext_vector_type(8))) float v8f;

// ---------------- transpose weights: Wt[sel][n][k] = W[sel][k][n] ------------
__global__ void gat_wt_kernel(const float* __restrict__ Wq,
                              const float* __restrict__ Wk,
                              const float* __restrict__ Wv,
                              float* __restrict__ Wt) {
    int i = blockIdx.x * blockDim.x + threadIdx.x;     // 0 .. 3*128*128-1
    if (i >= 3 * IN_DIM * QKV_DIM) return;
    const int sel = i / (IN_DIM * QKV_DIM);
    const int r   = i % (IN_DIM * QKV_DIM);
    const int n   = r >> 7;          // output column
    const int k   = r & 127;         // input dim
    const float* W = (sel == 0) ? Wq : (sel == 1) ? Wk : Wv;
    Wt[i] = W[k * QKV_DIM + n];
}

// ---------------- zero accumulators (float4) ----------------
__global__ void gat_zero_kernel(float4* __restrict__ wV, float* __restrict__ z,
                                int nOut4, int nZ) {
    int i = blockIdx.x * blockDim.x + threadIdx.x;
    if (i < nOut4) wV[i] = make_float4(0.f, 0.f, 0.f, 0.f);
    if (i < nZ)    z[i]  = 0.f;
}

// ---------------- fused Q/K/V projection via V_WMMA_F32_16X16X4_F32 ----------
// One wave32 computes a 16-node x 64-col strip (4 accumulators) so each A
// fragment feeds 4 WMMAs. waveTile 0..5: {Q lo, Q hi, K lo, K hi, V lo, V hi}.
__global__ __launch_bounds__(64) void gat_proj_kernel(
    const float* __restrict__ h, const float* __restrict__ Wt,
    const float* __restrict__ bq, const float* __restrict__ bk,
    const float* __restrict__ bv,
    float* __restrict__ Q, float* __restrict__ K, float* __restrict__ V,
    int nNodes) {
    const int lane     = threadIdx.x;                    // 0..31
    const int waveTile = blockIdx.y * 2 + threadIdx.y;   // 0..5
    const int nodeBase = blockIdx.x * 16;
    const int sel      = waveTile >> 1;                  // 0=Q 1=K 2=V
    const int colBase0 = (waveTile & 1) * 64;

    const float* __restrict__ b = (sel == 0) ? bq : (sel == 1) ? bk : bv;
    float* __restrict__ out     = (sel == 0) ? Q  : (sel == 1) ? K  : V;
    const float* __restrict__ W = Wt + (size_t)sel * IN_DIM * QKV_DIM; // [n][k]

    const int mr = lane & 15;            // A: row M ; B: col N (within tile)
    const int kb = (lane >> 4) * 2;      // K sub-offset per ISA f32 layout

    const float* __restrict__ arow = h + (size_t)(nodeBase + mr) * IN_DIM;
    const float* __restrict__ br0  = W + (size_t)(colBase0 + mr) * IN_DIM;
    const float* __restrict__ br1  = br0 + 16 * IN_DIM;
    const float* __restrict__ br2  = br0 + 32 * IN_DIM;
    const float* __restrict__ br3  = br0 + 48 * IN_DIM;

    v8f acc0 = {}, acc1 = {}, acc2 = {}, acc3 = {};
#pragma unroll
    for (int k0 = 0; k0 < IN_DIM; k0 += 4) {
        const int kk = k0 + kb;
        v2f a, b0, b1, b2, b3;
        a.x  = arow[kk]; a.y  = arow[kk + 1];
        b0.x = br0[kk];  b0.y = br0[kk + 1];      // contiguous -> b64 load
        b1.x = br1[kk];  b1.y = br1[kk + 1];
        b2.x = br2[kk];  b2.y = br2[kk + 1];
        b3.x = br3[kk];  b3.y = br3[kk + 1];
        acc0 = __builtin_amdgcn_wmma_f32_16x16x4_f32(false, a, false, b0,
                                                     (short)0, acc0, false, false);
        acc1 = __builtin_amdgcn_wmma_f32_16x16x4_f32(false, a, false, b1,
                                                     (short)0, acc1, false, false);
        acc2 = __builtin_amdgcn_wmma_f32_16x16x4_f32(false, a, false, b2,
                                                     (short)0, acc2, false, false);
        acc3 = __builtin_amdgcn_wmma_f32_16x16x4_f32(false, a, false, b3,
                                                     (short)0, acc3, false, false);
    }

#pragma unroll
    for (int j = 0; j < 4; ++j) {
        const v8f acc = (j == 0) ? acc0 : (j == 1) ? acc1 : (j == 2) ? acc2 : acc3;
        const int colBase = colBase0 + 16 * j;
        const float bias  = b[colBase + mr];
#pragma unroll
        for (int v = 0; v < 8; ++v) {
            const int row = v + (lane >> 4) * 8;   // C/D layout
            out[(size_t)(nodeBase + row) * QKV_DIM + colBase + mr] = acc[v] + bias;
        }
    }
}

// ---------------- edge phase: one wave32 per edge, float4 loads --------------
// lane -> head = lane>>2, quad = lane&3 covers dims quad*4..quad*4+3.
__global__ __launch_bounds__(256) void gat_edge_kernel(
    const float* __restrict__ Q, const float* __restrict__ K,
    const float* __restrict__ V,
    const int* __restrict__ src, const int* __restrict__ dst,
    float* __restrict__ wV, float* __restrict__ z, int nEdges) {
    const int gid  = blockIdx.x * blockDim.x + threadIdx.x;
    const int edge = gid >> 5;
    if (edge >= nEdges) return;
    const int lane = threadIdx.x & 31;
    const int head = lane >> 2;
    const int quad = lane & 3;
    const int base = head * HEAD_DIM + quad * 4;

    const int s = src[edge];
    const int d = dst[edge];

    const float4 k4 = *(const float4*)(K + (size_t)s * QKV_DIM + base);
    const float4 q4 = *(const float4*)(Q + (size_t)d * QKV_DIM + base);
    float p = k4.x * q4.x + k4.y * q4.y + k4.z * q4.z + k4.w * q4.w;
    // per-head dot: reduce across the 4-lane quad group
    p += __shfl_xor(p, 1, 32);
    p += __shfl_xor(p, 2, 32);

    float sc = p * 0.25f;                           // / sqrt(16)
    sc = fminf(5.0f, fmaxf(-5.0f, sc));
    sc = __expf(sc);

    const float4 v4 = *(const float4*)(V + (size_t)s * QKV_DIM + base);
    float* o = wV + (size_t)d * QKV_DIM + base;
    atomicAdd(o + 0, v4.x * sc);
    atomicAdd(o + 1, v4.y * sc);
    atomicAdd(o + 2, v4.z * sc);
    atomicAdd(o + 3, v4.w * sc);
    if (quad == 0)
        atomicAdd(&z[(size_t)d * N_HEADS + head], sc);
}

// ---------------- normalize: out = wV / (z + 1e-6), float4 in place ----------
__global__ void gat_norm_kernel(float4* __restrict__ wV,
                                const float* __restrict__ z, int nOut4) {
    int i = blockIdx.x * blockDim.x + threadIdx.x;
    if (i >= nOut4) return;
    const int node = i >> 5;             // (i*4)/128
    const int head = (i >> 2) & 7;
    const float den = z[node * N_HEADS + head] + 1e-6f;
    float4 v = wV[i];
    v.x = v.x / den; v.y = v.y / den; v.z = v.z / den; v.w = v.w / den;
    wV[i] = v;
}

extern "C" void kernel_launch(void* const* d_in, const int* in_sizes, int n_in,
                              void* d_out, int out_size, void* d_ws, size_t ws_size,
                              hipStream_t stream) {
    const float* h  = (const float*)d_in[0];
    const float* Wq = (const float*)d_in[1];
    const float* bq = (const float*)d_in[2];
    const float* Wk = (const float*)d_in[3];
    const float* bk = (const float*)d_in[4];
    const float* Wv = (const float*)d_in[5];
    const float* bv = (const float*)d_in[6];
    const int*  src = (const int*)d_in[7];
    const int*  dst = (const int*)d_in[8];
    float* out = (float*)d_out;

    const int nNodes = in_sizes[0] / IN_DIM;
    const int nEdges = in_sizes[7];
    const int nOut   = nNodes * QKV_DIM;    // 6.4M floats
    const int nOut4  = nOut / 4;
    const int nZ     = nNodes * N_HEADS;

    // workspace: Q | K | V | z | Wt   (fp32)
    float* Q  = (float*)d_ws;
    float* K  = Q + (size_t)nOut;
    float* V  = K + (size_t)nOut;
    float* z  = V + (size_t)nOut;
    float* Wt = z + (size_t)nZ;

    // 0) transpose weights (tiny: 196KB)
    gat_wt_kernel<<<(3 * IN_DIM * QKV_DIM + 255) / 256, 256, 0, stream>>>(
        Wq, Wk, Wv, Wt);

    // 1) zero accumulators (d_out is the wV accumulator)
    gat_zero_kernel<<<(nOut4 + 255) / 256, 256, 0, stream>>>(
        (float4*)out, z, nOut4, nZ);

    // 2) fused Q/K/V projection (WMMA f32 16x16x4, 4 tiles per wave)
    dim3 pgrid((nNodes + 15) / 16, 3);
    dim3 pblock(32, 2);
    gat_proj_kernel<<<pgrid, pblock, 0, stream>>>(h, Wt, bq, bk, bv,
                                                  Q, K, V, nNodes);

    // 3) edge scatter phase (one wave32 per edge, 8 edges per block)
    gat_edge_kernel<<<(nEdges + 7) / 8, 256, 0, stream>>>(Q, K, V, src, dst,
                                                          out, z, nEdges);

    // 4) normalize in place
    gat_norm_kernel<<<(nOut4 + 255) / 256, 256, 0, stream>>>(
        (float4*)out, z, nOut4);
}